// Linear_33500744909353
// MI455X (gfx1250) — compile-verified
//
#include <hip/hip_runtime.h>
#include <hip/hip_bf16.h>

typedef __attribute__((ext_vector_type(2))) float v2f;
typedef __attribute__((ext_vector_type(8))) float v8f;
typedef __attribute__((ext_vector_type(4))) unsigned int u32x4;
typedef __attribute__((ext_vector_type(8))) int i32x8;
typedef __attribute__((ext_vector_type(4))) int i32x4;

#define NCLASSES 8
#define NSIZE    512
#define NINPUT   512
#define NCTX     512
#define NBATCH   64
#define NCMS     4
#define NWROWS   16   // 2^CMS weight rows per (c,s)
#define LRATE    0.01f
#define WCLIP    5.0f
// logit(0.01) = log(0.01) - log1p(-0.01)
#define PCLIP_LO (-4.59511985f)
#define PCLIP_HI ( 4.59511985f)

__device__ __forceinline__ v8f wmma_f32(v2f a, v2f b, v8f c) {
  // D = A(16x4,f32) * B(4x16,f32) + C(16x16,f32)
  return __builtin_amdgcn_wmma_f32_16x16x4_f32(
      /*neg_a=*/false, a, /*neg_b=*/false, b,
      /*c_mod=*/(short)0, c, /*reuse_a=*/false, /*reuse_b=*/false);
}

// ---------------------------------------------------------------------------
// Kernel A: distances = einsum('cskd,bd->cskb') via V_WMMA_F32_16X16X4_F32,
// then threshold against bias and pack the 4 context bits into idx[c][s][b].
// Grid: (16, C) blocks of 256 threads (8 waves). Each wave owns a 16-row
// M-tile of the flattened (s,k) dimension (2048 rows per class) x all 64 b.
// ---------------------------------------------------------------------------
__global__ __launch_bounds__(256) void gln_ctx_idx(
    const float* __restrict__ cmaps,    // (C, S, CMS, CTX)
    const float* __restrict__ context,  // (B, CTX)
    const float* __restrict__ cbias,    // (C, S, CMS, 1)
    unsigned char* __restrict__ idx_out)// (C, S, B)
{
  const int lane = threadIdx.x & 31;
  const int wave = threadIdx.x >> 5;
  const int ln   = lane & 15;
  const int hi   = lane >> 4;          // 0: K=0,1  1: K=2,3 (fp32 WMMA layout)
  const int c    = blockIdx.y;
  const int mb   = blockIdx.x * 128 + wave * 16;   // M-tile base in [0,2048)

  const float* arow = cmaps + ((size_t)(c * 2048 + mb + ln)) * NCTX + 2 * hi;
  const float* bp0  = context + (size_t)( 0 + ln) * NCTX + 2 * hi;
  const float* bp1  = context + (size_t)(16 + ln) * NCTX + 2 * hi;
  const float* bp2  = context + (size_t)(32 + ln) * NCTX + 2 * hi;
  const float* bp3  = context + (size_t)(48 + ln) * NCTX + 2 * hi;

  v8f acc0 = {}; v8f acc1 = {}; v8f acc2 = {}; v8f acc3 = {};
  for (int d = 0; d < NCTX; d += 4) {
    v2f a  = *(const v2f*)(arow + d);
    v2f b0 = *(const v2f*)(bp0 + d);
    v2f b1 = *(const v2f*)(bp1 + d);
    v2f b2 = *(const v2f*)(bp2 + d);
    v2f b3 = *(const v2f*)(bp3 + d);
    acc0 = wmma_f32(a, b0, acc0);
    acc1 = wmma_f32(a, b1, acc1);
    acc2 = wmma_f32(a, b2, acc2);
    acc3 = wmma_f32(a, b3, acc3);
  }

  // This lane owns rows m = mb + 8*hi + r (r=0..7) => two s values, k = r%4.
  float bias[8];
#pragma unroll
  for (int r = 0; r < 8; ++r)
    bias[r] = cbias[c * 2048 + mb + 8 * hi + r];
  const int s0 = (mb >> 2) + 2 * hi;

  {
    v8f accs[4] = {acc0, acc1, acc2, acc3};
#pragma unroll
    for (int nt = 0; nt < 4; ++nt) {
      const int b = nt * 16 + ln;
      int i0 = 0, i1 = 0;
#pragma unroll
      for (int r = 0; r < 4; ++r) {
        i0 |= (accs[nt][r]     > bias[r])     ? (1 << r) : 0;
        i1 |= (accs[nt][4 + r] > bias[4 + r]) ? (1 << r) : 0;
      }
      idx_out[(size_t)(c * NSIZE + s0    ) * NBATCH + b] = (unsigned char)i0;
      idx_out[(size_t)(c * NSIZE + s0 + 1) * NBATCH + b] = (unsigned char)i1;
    }
  }
}

// ---------------------------------------------------------------------------
// Kernel B: one (c,s) pair per 64-thread block (2 waves).
//  1. TDM tensor_load_to_lds stages the 16x512 weight tile (32 KB) in LDS:
//     1-D descriptor, data_size=4B, tile_dim0=8192, tracked by TENSORcnt.
//  2. P = W(16x512) @ logits_c(512x64) with fp32 WMMA (A-frags from LDS).
//  3. out[b] = clip(P[idx_b, b]); sigmoid; diff.
//  4. Scatter semantics: last duplicate b wins a bucket; update winner rows
//     with clip(W - LR*diff*logits[:,b], +-5); copy-through untouched rows.
// Weights stream: 134 MB in + 134 MB out = the roofline minimum.
// ---------------------------------------------------------------------------
__global__ __launch_bounds__(64) void gln_predict_update(
    const float* __restrict__ logits,   // (C, I, B)
    const float* __restrict__ target,   // (C, B)
    const float* __restrict__ weights,  // (C, S, 16, I)
    const unsigned char* __restrict__ idx, // (C, S, B)
    float* __restrict__ out,            // (C, S, B)
    float* __restrict__ wout)           // (C, S, 16, I)
{
  __shared__ float         Wl[NWROWS * NINPUT];   // 32 KB weight tile
  __shared__ float         Pl[NWROWS * NBATCH];   // 4 KB prediction tile
  __shared__ float         diffl[NBATCH];
  __shared__ int           winl[NWROWS];
  __shared__ float         dwl[NWROWS];
  __shared__ unsigned char idxl[NBATCH];

  const int tid  = threadIdx.x;
  const int wave = tid >> 5;
  const int lane = tid & 31;
  const int ln   = lane & 15;
  const int hi   = lane >> 4;
  const int s    = blockIdx.x & (NSIZE - 1);
  const int c    = blockIdx.x >> 9;

  // --- stage W tile via Tensor Data Mover (wave 0 issues; TDM ignores EXEC,
  //     so wave 1 branches around and synchronizes at the barrier) ---
  if (tid < 32) {
    const unsigned long long gaddr =
        (unsigned long long)(const void*)(weights +
            (size_t)(c * NSIZE + s) * (NWROWS * NINPUT));
    const unsigned int ldsb = (unsigned int)(unsigned long long)(const void*)&Wl[0];
    const unsigned int nelem = NWROWS * NINPUT;   // 8192 fp32 elements, 1-D tile
    u32x4 g0;
    g0.x = 1u;                                    // count=1, user descriptor
    g0.y = ldsb;                                  // lds_addr[31:0]
    g0.z = (unsigned int)gaddr;                   // global_addr[31:0]
    g0.w = (unsigned int)((gaddr >> 32) & 0x01FFFFFFull) | (2u << 30); // type=2
    i32x8 g1;
    g1[0] = (int)(2u << 16);                      // data_size=2 (4 bytes)
    g1[1] = (int)((nelem & 0xFFFFu) << 16);       // tensor_dim0[15:0]
    g1[2] = (int)((nelem >> 16) & 0xFFFFu);       // tensor_dim0[31:16]
    g1[3] = (int)((nelem & 0xFFFFu) << 16);       // tile_dim0
    g1[4] = 0; g1[5] = 0; g1[6] = 0; g1[7] = 0;
    i32x4 gz  = {0, 0, 0, 0};
    i32x8 gz8 = {0, 0, 0, 0, 0, 0, 0, 0};
    __builtin_amdgcn_tensor_load_to_lds(g0, g1, gz, gz, gz8, 0);
    __builtin_amdgcn_s_wait_tensorcnt(0);
  }
  __syncthreads();

  // --- P = W @ logits_c ; wave owns two 16-wide N tiles ---
  const float* Lg  = logits + (size_t)c * NINPUT * NBATCH;
  const int    nt0 = wave * 2, nt1 = nt0 + 1;
  v8f acc0 = {}; v8f acc1 = {};
  for (int d = 0; d < NINPUT; d += 4) {
    v2f a = *(const v2f*)&Wl[ln * NINPUT + d + 2 * hi];
    const float* l0 = Lg + (size_t)(d + 2 * hi) * NBATCH;
    v2f b0 = { l0[nt0 * 16 + ln], l0[NBATCH + nt0 * 16 + ln] };
    v2f b1 = { l0[nt1 * 16 + ln], l0[NBATCH + nt1 * 16 + ln] };
    acc0 = wmma_f32(a, b0, acc0);
    acc1 = wmma_f32(a, b1, acc1);
  }
#pragma unroll
  for (int r = 0; r < 8; ++r) {
    const int j = r + 8 * hi;            // C/D layout: VGPR r, lane-half hi
    Pl[j * NBATCH + nt0 * 16 + ln] = acc0[r];
    Pl[j * NBATCH + nt1 * 16 + ln] = acc1[r];
  }
  __syncthreads();

  // --- select, clip, sigmoid, diff (one b per thread) ---
  {
    const int b = tid;
    const unsigned char j = idx[(size_t)(c * NSIZE + s) * NBATCH + b];
    idxl[b] = j;
    float p = Pl[(int)j * NBATCH + b];
    p = fminf(fmaxf(p, PCLIP_LO), PCLIP_HI);
    out[(size_t)(c * NSIZE + s) * NBATCH + b] = p;
    const float sig = 1.0f / (1.0f + __expf(-p));
    diffl[b] = sig - target[c * NBATCH + b];
  }
  __syncthreads();

  // --- per-bucket winner (last duplicate index wins the scatter) ---
  if (tid < NWROWS) {
    int w = -1;
    for (int b = 0; b < NBATCH; ++b)
      if (idxl[b] == (unsigned char)tid) w = b;
    winl[tid] = w;
    dwl[tid]  = (w >= 0) ? LRATE * diffl[w] : 0.0f;
  }
  __syncthreads();

  // --- update winners, copy-through others; coalesced row writes ---
  float* wo = wout + (size_t)(c * NSIZE + s) * (NWROWS * NINPUT);
#pragma unroll 1
  for (int j = 0; j < NWROWS; ++j) {
    const int   w  = winl[j];
    const float dv = dwl[j];
#pragma unroll
    for (int t = 0; t < NINPUT / 64; ++t) {
      const int i = tid + t * 64;
      float wv = Wl[j * NINPUT + i];
      if (w >= 0) {
        wv = wv - dv * Lg[(size_t)i * NBATCH + w];
        wv = fminf(fmaxf(wv, -WCLIP), WCLIP);
      }
      wo[j * NINPUT + i] = wv;
    }
  }
}

extern "C" void kernel_launch(void* const* d_in, const int* in_sizes, int n_in,
                              void* d_out, int out_size, void* d_ws, size_t ws_size,
                              hipStream_t stream) {
  const float* logits  = (const float*)d_in[0]; // (C, I, B)
  const float* context = (const float*)d_in[1]; // (B, D)
  const float* target  = (const float*)d_in[2]; // (C, B)
  const float* cmaps   = (const float*)d_in[3]; // (C, S, CMS, D)
  const float* cbias   = (const float*)d_in[4]; // (C, S, CMS, 1)
  const float* weights = (const float*)d_in[5]; // (C, S, 16, I)

  float* out  = (float*)d_out;                               // (C,S,B)
  float* wout = out + (size_t)NCLASSES * NSIZE * NBATCH;     // (C,S,16,I)
  unsigned char* idx = (unsigned char*)d_ws;                 // (C,S,B) = 256 KB

  gln_ctx_idx<<<dim3(16, NCLASSES), 256, 0, stream>>>(cmaps, context, cbias, idx);
  gln_predict_update<<<dim3(NCLASSES * NSIZE), 64, 0, stream>>>(
      logits, target, weights, idx, out, wout);
}